// poseHead_43138651521629
// MI455X (gfx1250) — compile-verified
//
#include <hip/hip_runtime.h>
#include <math.h>

typedef __attribute__((ext_vector_type(2))) float v2f;
typedef __attribute__((ext_vector_type(4))) float v4f;
typedef __attribute__((ext_vector_type(8))) float v8f;

#define IMG_W 320
#define IMG_H 240
#define HW (IMG_W * IMG_H)   // 76800
#define NB 32
#define NC 8
#define NPROB (NB * NC)      // 256
#define SQRT2F 1.41421356237309504880f

// ---------------------------------------------------------------------------
// small matrix helpers (per-lane scalar)
// ---------------------------------------------------------------------------
__device__ inline float dot3(const float* a, const float* b) {
  return a[0]*b[0] + a[1]*b[1] + a[2]*b[2];
}
__device__ inline void cross3(const float* a, const float* b, float* c) {
  c[0] = a[1]*b[2] - a[2]*b[1];
  c[1] = a[2]*b[0] - a[0]*b[2];
  c[2] = a[0]*b[1] - a[1]*b[0];
}
__device__ inline void mat3_vec(const float* A, const float* x, float* y) {
  y[0] = A[0]*x[0] + A[1]*x[1] + A[2]*x[2];
  y[1] = A[3]*x[0] + A[4]*x[1] + A[5]*x[2];
  y[2] = A[6]*x[0] + A[7]*x[1] + A[8]*x[2];
}
__device__ inline void mat3_mul(const float* A, const float* B, float* C) {
  for (int i = 0; i < 3; ++i)
    for (int j = 0; j < 3; ++j)
      C[i*3+j] = A[i*3+0]*B[0*3+j] + A[i*3+1]*B[1*3+j] + A[i*3+2]*B[2*3+j];
}
__device__ inline void mat3_mul_At(const float* A, const float* B, float* C) { // C = A^T B
  for (int i = 0; i < 3; ++i)
    for (int j = 0; j < 3; ++j)
      C[i*3+j] = A[0*3+i]*B[0*3+j] + A[1*3+i]*B[1*3+j] + A[2*3+i]*B[2*3+j];
}
__device__ inline void inv3(const float* A, float* Ai) {
  float a=A[0],b=A[1],c=A[2],d=A[3],e=A[4],f=A[5],g=A[6],h=A[7],i=A[8];
  float det = a*(e*i - f*h) - b*(d*i - f*g) + c*(d*h - e*g);
  float r = 1.f / det;
  Ai[0]=(e*i-f*h)*r; Ai[1]=(c*h-b*i)*r; Ai[2]=(b*f-c*e)*r;
  Ai[3]=(f*g-d*i)*r; Ai[4]=(a*i-c*g)*r; Ai[5]=(c*d-a*f)*r;
  Ai[6]=(d*h-e*g)*r; Ai[7]=(b*g-a*h)*r; Ai[8]=(a*e-b*d)*r;
}

// cyclic Jacobi eigen-decomposition of symmetric NxN; A -> diag(eigvals), V cols = eigvecs
template <int N>
__device__ void jacobi_sym(float* A, float* V, int sweeps) {
  for (int i = 0; i < N*N; ++i) V[i] = 0.f;
  for (int i = 0; i < N; ++i) V[i*N+i] = 1.f;
  for (int s = 0; s < sweeps; ++s) {
    for (int p = 0; p < N-1; ++p) {
      for (int q = p+1; q < N; ++q) {
        float apq = A[p*N+q];
        float app = A[p*N+p], aqq = A[q*N+q];
        if (fabsf(apq) <= 1e-14f * (fabsf(app) + fabsf(aqq)) + 1e-32f) continue;
        float tau = (aqq - app) / (2.f * apq);
        float tt  = (tau >= 0.f ? 1.f : -1.f) / (fabsf(tau) + sqrtf(1.f + tau*tau));
        float c   = rsqrtf(1.f + tt*tt);
        float sn  = tt * c;
        for (int k = 0; k < N; ++k) {
          float akp = A[k*N+p], akq = A[k*N+q];
          A[k*N+p] = c*akp - sn*akq;
          A[k*N+q] = sn*akp + c*akq;
        }
        for (int k = 0; k < N; ++k) {
          float apk = A[p*N+k], aqk = A[q*N+k];
          A[p*N+k] = c*apk - sn*aqk;
          A[q*N+k] = sn*apk + c*aqk;
        }
        for (int k = 0; k < N; ++k) {
          float vkp = V[k*N+p], vkq = V[k*N+q];
          V[k*N+p] = c*vkp - sn*vkq;
          V[k*N+q] = sn*vkp + c*vkq;
        }
      }
    }
  }
}

// 3x3 SVD (U columns, Vh rows); rank-aware: u2 = u0 x u1 so sigma3~0 is safe.
__device__ void svd3_UVh(const float* F, float* U, float* Vh) {
  float G[9];
  for (int i = 0; i < 3; ++i)
    for (int j = 0; j < 3; ++j)
      G[i*3+j] = F[0*3+i]*F[0*3+j] + F[1*3+i]*F[1*3+j] + F[2*3+i]*F[2*3+j];
  float V[9];
  jacobi_sym<3>(G, V, 10);
  float lam[3] = {G[0], G[4], G[8]};
  int o[3] = {0, 1, 2};
  for (int a = 0; a < 2; ++a)
    for (int b = a+1; b < 3; ++b)
      if (lam[o[b]] > lam[o[a]]) { int t = o[a]; o[a] = o[b]; o[b] = t; }
  float v0[3], v1[3], v2[3];
  for (int r = 0; r < 3; ++r) { v0[r]=V[r*3+o[0]]; v1[r]=V[r*3+o[1]]; v2[r]=V[r*3+o[2]]; }
  float u0[3], u1[3], u2[3];
  mat3_vec(F, v0, u0);
  float n0 = rsqrtf(fmaxf(dot3(u0,u0), 1e-30f));
  for (int r = 0; r < 3; ++r) u0[r] *= n0;
  mat3_vec(F, v1, u1);
  float pr = dot3(u0, u1);
  for (int r = 0; r < 3; ++r) u1[r] -= pr * u0[r];
  float n1 = rsqrtf(fmaxf(dot3(u1,u1), 1e-30f));
  for (int r = 0; r < 3; ++r) u1[r] *= n1;
  cross3(u0, u1, u2);
  for (int r = 0; r < 3; ++r) { U[r*3+0]=u0[r]; U[r*3+1]=u1[r]; U[r*3+2]=u2[r]; }
  for (int c = 0; c < 3; ++c) { Vh[0*3+c]=v0[c]; Vh[1*3+c]=v1[c]; Vh[2*3+c]=v2[c]; }
}

__device__ void hartley(const float* P, float* Pn, float* T) {
  float mx = 0.f, my = 0.f;
  for (int i = 0; i < 8; ++i) { mx += P[2*i]; my += P[2*i+1]; }
  mx *= 0.125f; my *= 0.125f;
  float mn = 0.f;
  for (int i = 0; i < 8; ++i) {
    float dx = P[2*i] - mx, dy = P[2*i+1] - my;
    mn += sqrtf(dx*dx + dy*dy);
  }
  mn *= 0.125f;
  float s = SQRT2F / mn;
  T[0]=s;   T[1]=0.f; T[2]=-s*mx;
  T[3]=0.f; T[4]=s;   T[5]=-s*my;
  T[6]=0.f; T[7]=0.f; T[8]=1.f;
  float iz = 1.f / (1.f + 1e-10f);   // de_homo's +1e-10 with z==1
  for (int i = 0; i < 8; ++i) {
    Pn[2*i]   = s*(P[2*i]   - mx) * iz;
    Pn[2*i+1] = s*(P[2*i+1] - my) * iz;
  }
}

__device__ int cheirality8(const float* X, const float* Y, const float* R, const float* t) {
  int cnt = 0;
  for (int i = 0; i < 8; ++i) {
    float Xh[3] = {X[2*i], X[2*i+1], 1.f};
    float Yh[3] = {Y[2*i], Y[2*i+1], 1.f};
    float Yc[3]; mat3_vec(R, Yh, Yc);
    float cr[3]; cross3(Xh, Yc, cr);
    float d = t[0]*cr[0] + t[1]*cr[1] + t[2]*cr[2];
    cnt += (d > 0.f) ? 1 : 0;
  }
  return cnt;
}

// ---------------------------------------------------------------------------
// Kernel 1: masks = (motion == channel-max).  Streaming, HBM-bound.
// Regular loads keep motion_map resident in the 192MB L2 for the top-k pass;
// non-temporal stores keep the 78MB write stream from evicting it.
// ---------------------------------------------------------------------------
__global__ __launch_bounds__(256) void masks_kernel(const float* __restrict__ motion,
                                                    float* __restrict__ masks_out) {
  const int quads = HW / 4;
  int tid = blockIdx.x * 256 + threadIdx.x;
  if (tid >= NB * quads) return;
  int b = tid / quads;
  int q = tid - b * quads;
  const v4f* mp = (const v4f*)motion;
  v4f v[NC];
#pragma unroll
  for (int c = 0; c < NC; ++c)
    v[c] = mp[(size_t)(b*NC + c) * quads + q];
  v4f mx = v[0];
#pragma unroll
  for (int c = 1; c < NC; ++c) {
    mx.x = fmaxf(mx.x, v[c].x); mx.y = fmaxf(mx.y, v[c].y);
    mx.z = fmaxf(mx.z, v[c].z); mx.w = fmaxf(mx.w, v[c].w);
  }
  v4f* op = (v4f*)masks_out;
#pragma unroll
  for (int c = 0; c < NC; ++c) {
    v4f o;
    o.x = (v[c].x == mx.x) ? 1.f : 0.f;
    o.y = (v[c].y == mx.y) ? 1.f : 0.f;
    o.z = (v[c].z == mx.z) ? 1.f : 0.f;
    o.w = (v[c].w == mx.w) ? 1.f : 0.f;
    __builtin_nontemporal_store(o, &op[(size_t)(b*NC + c) * quads + q]);
  }
}

// ---------------------------------------------------------------------------
// Kernel 2: per-(b,c) top-8 of motion (value desc, tie -> lower index) and
// gather of coords / flow / depth into workspace (40 floats per problem).
// One 256-thread block per problem; motion re-read hits L2.
// ---------------------------------------------------------------------------
__global__ __launch_bounds__(256) void topk_kernel(const float* __restrict__ motion,
                                                   const float* __restrict__ flow,
                                                   const float* __restrict__ depth,
                                                   float* __restrict__ ws) {
  __shared__ float sv[256*8];
  __shared__ int   si[256*8];
  const int p = blockIdx.x;          // b*NC + c
  const int b = p / NC;
  const float* m = motion + (size_t)p * HW;

  float v[8]; int id[8];
#pragma unroll
  for (int k = 0; k < 8; ++k) { v[k] = -3.4e38f; id[k] = 0x7fffffff; }
  for (int i = threadIdx.x; i < HW; i += 256) {
    float x = m[i];
    if (x > v[7]) {
      int j = 7;
      while (j > 0 && x > v[j-1]) { v[j] = v[j-1]; id[j] = id[j-1]; --j; }
      v[j] = x; id[j] = i;
    }
  }
#pragma unroll
  for (int k = 0; k < 8; ++k) { sv[threadIdx.x*8+k] = v[k]; si[threadIdx.x*8+k] = id[k]; }
  __syncthreads();

  if (threadIdx.x == 0) {
    float* o = ws + (size_t)p * 40;
    for (int k = 0; k < 8; ++k) {
      float best = -3.4e38f; int bidx = 0x7fffffff; int bloc = 0;
      for (int j = 0; j < 256*8; ++j) {
        float vv = sv[j]; int ii = si[j];
        if (vv > best || (vv == best && ii < bidx)) { best = vv; bidx = ii; bloc = j; }
      }
      sv[bloc] = -3.4e38f;
      int y = bidx / IMG_W;
      int x = bidx - y * IMG_W;
      float f0 = flow[((size_t)b*2 + 0) * HW + bidx];
      float f1 = flow[((size_t)b*2 + 1) * HW + bidx];
      float d  = depth[(size_t)b * HW + bidx];
      o[2*k+0]      = (float)y;            // coords[...,0] = y (reference quirk)
      o[2*k+1]      = (float)x;
      o[16 + 2*k+0] = (float)y + f0;
      o[16 + 2*k+1] = (float)x + f1;
      o[32 + k]     = d;
    }
  }
}

// ---------------------------------------------------------------------------
// Kernel 3: 8-point pose solve.  One wave (32 lanes) per problem.
// The 9x9 normal matrix M = XX^T * XX is formed with two
// V_WMMA_F32_16X16X4_F32 (fp32 matrix pipe, K=8 split into 2x K=4);
// A = XX^T and B = XX share identical per-lane operand layouts (M symmetric),
// so one v2f register pair feeds both SRC0 and SRC1.
// Everything else (Jacobi eigen, 3x3 SVDs, cheirality, triangulation) is
// lane-redundant scalar VALU — negligible vs the streaming kernels.
// ---------------------------------------------------------------------------
__global__ __launch_bounds__(32) void solve_kernel(const float* __restrict__ Kmat,
                                                   const float* __restrict__ ws,
                                                   float* __restrict__ Tout) {
  __shared__ float sXX[8*9];
  __shared__ float sM[16*16];
  __shared__ float sRatio[8];
  const int p    = blockIdx.x;
  const int lane = threadIdx.x;
  const float* in = ws + (size_t)p * 40;

  float X[16], Y[16], dsel[8];
#pragma unroll
  for (int i = 0; i < 8; ++i) {
    X[2*i]   = in[2*i];      X[2*i+1] = in[2*i+1];
    Y[2*i]   = in[16+2*i];   Y[2*i+1] = in[16+2*i+1];
    dsel[i]  = in[32+i];
  }
  float K[9], Kinv[9];
#pragma unroll
  for (int i = 0; i < 9; ++i) K[i] = Kmat[i];
  inv3(K, Kinv);

  // pixel -> normalized camera coords (with reference's +1e-10 de_homo eps)
  float Xc[16], Yc[16];
  for (int i = 0; i < 8; ++i) {
    float h[3] = {X[2*i], X[2*i+1], 1.f}, r[3];
    mat3_vec(Kinv, h, r);
    float iz = 1.f / (r[2] + 1e-10f);
    Xc[2*i] = r[0]*iz; Xc[2*i+1] = r[1]*iz;
    h[0] = Y[2*i]; h[1] = Y[2*i+1]; h[2] = 1.f;
    mat3_vec(Kinv, h, r);
    iz = 1.f / (r[2] + 1e-10f);
    Yc[2*i] = r[0]*iz; Yc[2*i+1] = r[1]*iz;
  }
  float T1[9], T2[9], Xn[16], Yn[16];
  hartley(Xc, Xn, T1);
  hartley(Yc, Yn, T2);

  // build 8x9 measurement matrix XX
  float XXr[8][9];
  for (int i = 0; i < 8; ++i) {
    float x0 = Xn[2*i], x1 = Xn[2*i+1], x2 = Yn[2*i], x3 = Yn[2*i+1];
    XXr[i][0] = x2*x0; XXr[i][1] = x2*x1; XXr[i][2] = x2;
    XXr[i][3] = x3*x0; XXr[i][4] = x3*x1; XXr[i][5] = x3;
    XXr[i][6] = x0;    XXr[i][7] = x1;    XXr[i][8] = 1.f;
  }
  if (lane < 8) {
#pragma unroll
    for (int j = 0; j < 9; ++j) sXX[lane*9 + j] = XXr[lane][j];
  }
  __syncthreads();

  // ---- WMMA: M(16x16) = padded(XX^T)(16x8) x padded(XX)(8x16) ----
  {
    const int half = lane >> 4;       // 0: lanes 0-15, 1: lanes 16-31
    const int lm   = lane & 15;       // m for A / n for B
    const bool ok  = (lm < 9);
    // f32 A layout: lanes 0-15 hold K={0,1} in {v0,v1}; lanes 16-31 hold K={2,3}
    // f32 B layout: v0 holds K=2*half, v1 holds K=2*half+1 -> same index formula,
    // and A[m][k]=XX[k][m], B[k][n]=XX[k][n] with m=n=lm -> identical values.
    v2f ab0, ab1;
    ab0.x = ok ? sXX[(2*half + 0)*9 + lm] : 0.f;
    ab0.y = ok ? sXX[(2*half + 1)*9 + lm] : 0.f;
    ab1.x = ok ? sXX[(4 + 2*half + 0)*9 + lm] : 0.f;
    ab1.y = ok ? sXX[(4 + 2*half + 1)*9 + lm] : 0.f;
    v8f acc = {};
    acc = __builtin_amdgcn_wmma_f32_16x16x4_f32(false, ab0, false, ab0,
                                                (short)0, acc, false, false);
    acc = __builtin_amdgcn_wmma_f32_16x16x4_f32(false, ab1, false, ab1,
                                                (short)0, acc, false, false);
    union { v8f v; float f[8]; } u; u.v = acc;
#pragma unroll
    for (int r = 0; r < 8; ++r)
      sM[(r + 8*half)*16 + lm] = u.f[r];   // C layout: VGPR r -> row r (+8 for hi lanes)
  }
  __syncthreads();

  // smallest eigenvector of M (= last right singular vector of XX) -> F
  float M9[81], V9[81];
  for (int i = 0; i < 9; ++i)
    for (int j = 0; j < 9; ++j)
      M9[i*9+j] = sM[i*16 + j];
  jacobi_sym<9>(M9, V9, 12);
  int mi = 0; float mv = M9[0];
  for (int i = 1; i < 9; ++i) { float d = M9[i*9+i]; if (d < mv) { mv = d; mi = i; } }
  float F[9];
  for (int i = 0; i < 9; ++i) F[i] = V9[i*9 + mi];

  // project to essential manifold: E110 = U diag(1,1,0) Vh ; E = T2^T E110 T1
  float UF[9], VhF[9];
  svd3_UVh(F, UF, VhF);
  float E110[9];
  for (int i = 0; i < 3; ++i)
    for (int j = 0; j < 3; ++j)
      E110[i*3+j] = UF[i*3+0]*VhF[0*3+j] + UF[i*3+1]*VhF[1*3+j];
  float tmp9[9], E[9];
  mat3_mul(E110, T1, tmp9);
  mat3_mul_At(T2, tmp9, E);

  // decompose E -> (R1,R2,t1,t2); reference computes R = U * W * Vh^T
  float UE[9], VhE[9];
  svd3_UVh(E, UE, VhE);
  float Vt[9];
  for (int i = 0; i < 3; ++i)
    for (int j = 0; j < 3; ++j)
      Vt[i*3+j] = VhE[j*3+i];
  const float Wm[9]  = {0.f,-1.f,0.f,  1.f,0.f,0.f,  0.f,0.f,1.f};
  const float WmT[9] = {0.f, 1.f,0.f, -1.f,0.f,0.f,  0.f,0.f,1.f};
  float R1[9], R2[9], tA[9];
  mat3_mul(UE, Wm,  tA); mat3_mul(tA, Vt, R1);
  mat3_mul(UE, WmT, tA); mat3_mul(tA, Vt, R2);
  float t1v[3] = {UE[0*3+2], UE[1*3+2], UE[2*3+2]};
  float t2v[3] = {-t1v[0], -t1v[1], -t1v[2]};

  // cheirality selection, order (R1,t1),(R1,t2),(R2,t1),(R2,t2), strict >
  float bR[9], bT[3]; int bs = -1;
  {
    const float* Rs[4] = {R1, R1, R2, R2};
    const float* Ts[4] = {t1v, t2v, t1v, t2v};
    for (int c = 0; c < 4; ++c) {
      int sc = cheirality8(X, Y, Rs[c], Ts[c]);
      if (sc > bs) {
        bs = sc;
        for (int i = 0; i < 9; ++i) bR[i] = Rs[c][i];
        for (int i = 0; i < 3; ++i) bT[i] = Ts[c][i];
      }
    }
  }

  // triangulate depths: P1 = [K|0], P2 = K[R|t]; one point per lane (mod 8)
  float P1[12], P2[12], Rt[12];
  for (int i = 0; i < 3; ++i) {
    for (int j = 0; j < 3; ++j) { P1[i*4+j] = K[i*3+j]; Rt[i*4+j] = bR[i*3+j]; }
    P1[i*4+3] = 0.f; Rt[i*4+3] = bT[i];
  }
  for (int i = 0; i < 3; ++i)
    for (int j = 0; j < 4; ++j)
      P2[i*4+j] = K[i*3+0]*Rt[0*4+j] + K[i*3+1]*Rt[1*4+j] + K[i*3+2]*Rt[2*4+j];
  {
    int i = lane & 7;
    float Xh[3] = {X[2*i], X[2*i+1], 1.f};
    float Yh[3] = {Y[2*i], Y[2*i+1], 1.f};
    float A4[16];
    for (int j = 0; j < 4; ++j) {
      A4[0*4+j] = Xh[0]*P1[2*4+j] - P1[0*4+j];
      A4[1*4+j] = Xh[1]*P1[2*4+j] - P1[1*4+j];
      A4[2*4+j] = Yh[0]*P2[2*4+j] - P2[0*4+j];
      A4[3*4+j] = Yh[1]*P2[2*4+j] - P2[1*4+j];
    }
    float B4[16];
    for (int a = 0; a < 4; ++a)
      for (int c = 0; c < 4; ++c) {
        float s = 0.f;
        for (int k = 0; k < 4; ++k) s += A4[k*4+a]*A4[k*4+c];
        B4[a*4+c] = s;
      }
    float V4[16];
    jacobi_sym<4>(B4, V4, 10);
    int m4 = 0; float mv4 = B4[0];
    for (int a = 1; a < 4; ++a) { float d = B4[a*4+a]; if (d < mv4) { mv4 = d; m4 = a; } }
    float dep = V4[2*4+m4] / V4[3*4+m4];
    if (lane < 8) sRatio[lane] = dep / dsel[i];
  }
  __syncthreads();

  if (lane == 0) {
    float scale = 0.f;
    for (int i = 0; i < 8; ++i) scale += sRatio[i];
    scale *= 0.125f;
    float* o = Tout + (size_t)p * 16;
    for (int i = 0; i < 3; ++i) {
      o[i*4+0] = bR[i*3+0]; o[i*4+1] = bR[i*3+1]; o[i*4+2] = bR[i*3+2];
      o[i*4+3] = bT[i] * scale;
    }
    o[12] = 0.f; o[13] = 0.f; o[14] = 0.f; o[15] = 1.f;
  }
}

// ---------------------------------------------------------------------------
extern "C" void kernel_launch(void* const* d_in, const int* in_sizes, int n_in,
                              void* d_out, int out_size, void* d_ws, size_t ws_size,
                              hipStream_t stream) {
  (void)in_sizes; (void)n_in; (void)out_size; (void)ws_size;
  const float* K      = (const float*)d_in[0];
  const float* flow   = (const float*)d_in[1];
  const float* motion = (const float*)d_in[2];
  const float* depth  = (const float*)d_in[3];
  float* out       = (float*)d_out;
  float* T_out     = out;                 // (B,C,4,4) = 4096 floats
  float* masks_out = out + NPROB * 16;    // (B,C,H,W) as 0/1 floats
  float* ws        = (float*)d_ws;        // 256 problems * 40 floats

  int quads = NB * (HW / 4);
  masks_kernel<<<(quads + 255) / 256, 256, 0, stream>>>(motion, masks_out);
  topk_kernel <<<NPROB, 256, 0, stream>>>(motion, flow, depth, ws);
  solve_kernel<<<NPROB, 32, 0, stream>>>(K, ws, T_out);
}